// Attention_22402549416684
// MI455X (gfx1250) — compile-verified
//
#include <hip/hip_runtime.h>
#include <math.h>

typedef float v2f __attribute__((ext_vector_type(2)));
typedef float v4f __attribute__((ext_vector_type(4)));
typedef float v8f __attribute__((ext_vector_type(8)));

#define B_  32
#define L_  2048
#define EN_ 1024
#define ZH_ 1024

// ---------------------------------------------------------------------------
// Kernel 1: v[b,e] = sum_h hx[b,h] * W[h,e]
// GEMM M=32(B), K=1024(ZH), N=1024(EN) with V_WMMA_F32_16X16X4_F32.
// One wave per 16x16 output tile; 2 M-tiles x 64 N-tiles = 128 waves.
// A (16x4 f32) lane layout (ISA 7.12.2): lane m=l%16, VGPR0=K(2*(l/16)),
// VGPR1=K(2*(l/16)+1). B (4x16) mirrored: lane n=l%16, rows 2*(l/16)+{0,1}.
// D: VGPR i -> M = i + 8*(l/16), N = l%16.
// ---------------------------------------------------------------------------
__global__ __launch_bounds__(32) void proj_vec_wmma(
    const float* __restrict__ hx, const float* __restrict__ W,
    float* __restrict__ v)
{
  const int lane  = threadIdx.x;      // 0..31 (wave32)
  const int tile  = blockIdx.x;       // 0..127
  const int mtile = tile & 1;         // 2 tiles of 16 batch rows
  const int ntile = tile >> 1;        // 64 tiles of 16 EN cols
  const int half  = lane >> 4;        // 0 or 1
  const int mn    = lane & 15;

  const float* __restrict__ hxr = hx + (mtile * 16 + mn) * ZH_;  // A row
  const int bcol = ntile * 16 + mn;                              // B col

  v8f acc = {0.f, 0.f, 0.f, 0.f, 0.f, 0.f, 0.f, 0.f};

  #pragma unroll 4
  for (int k = 0; k < ZH_; k += 4) {
    const int kk = k + 2 * half;
    v2f a, b;
    a.x = hxr[kk];
    a.y = hxr[kk + 1];
    b.x = W[(size_t)(kk    ) * EN_ + bcol];
    b.y = W[(size_t)(kk + 1) * EN_ + bcol];
    // 8 args: (neg_a, A, neg_b, B, c_mod, C, reuse_a, reuse_b)
    acc = __builtin_amdgcn_wmma_f32_16x16x4_f32(
        false, a, false, b, (short)0, acc, false, false);
  }

  #pragma unroll
  for (int i = 0; i < 8; ++i) {
    const int brow = mtile * 16 + i + 8 * half;
    v[(size_t)brow * EN_ + ntile * 16 + mn] = acc[i];
  }
}

// ---------------------------------------------------------------------------
// Kernel 2: c[b] = hx[b] . bias   (tiny; one block per batch row)
// ---------------------------------------------------------------------------
__global__ __launch_bounds__(128) void bias_dot(
    const float* __restrict__ hx, const float* __restrict__ bias,
    float* __restrict__ c)
{
  __shared__ float red[128];
  const int b = blockIdx.x;
  float s = 0.f;
  for (int h = threadIdx.x; h < ZH_; h += 128)
    s += hx[(size_t)b * ZH_ + h] * bias[h];
  red[threadIdx.x] = s;
  __syncthreads();
  for (int off = 64; off > 0; off >>= 1) {
    if (threadIdx.x < off) red[threadIdx.x] += red[threadIdx.x + off];
    __syncthreads();
  }
  if (threadIdx.x == 0) c[b] = red[0];
}

// ---------------------------------------------------------------------------
// Kernel 3 (dominant, HBM-bound): energies[b,l] = enc[b,l,:] . v[b,:] + c[b]
// Streams 256 MB of enc exactly once with non-temporal b128 loads.
// Block = 256 threads (8 waves), 64 rows per block, v[b] staged in LDS.
// ---------------------------------------------------------------------------
__global__ __launch_bounds__(256) void energies_kernel(
    const float* __restrict__ enc, const float* __restrict__ v,
    const float* __restrict__ c, float* __restrict__ energ)
{
  __shared__ float vs[EN_];
  const int blocksPerB = L_ / 64;                 // 32
  const int b     = blockIdx.x / blocksPerB;
  const int lbase = (blockIdx.x % blocksPerB) * 64;

  // stage v[b] (4 KB) into LDS: 256 threads x one float4
  ((v4f*)vs)[threadIdx.x] = ((const v4f*)(v + (size_t)b * EN_))[threadIdx.x];
  __syncthreads();

  const float cb   = c[b];
  const int   wave = threadIdx.x >> 5;            // 0..7
  const int   lane = threadIdx.x & 31;

  #pragma unroll 2
  for (int ri = 0; ri < 8; ++ri) {
    const int l = lbase + wave * 8 + ri;
    const float* __restrict__ row = enc + ((size_t)b * L_ + l) * EN_;
    float sum = 0.f;
    #pragma unroll
    for (int t = 0; t < 8; ++t) {
      const int e = lane * 4 + t * 128;           // coalesced 512B per step
      const v4f d = __builtin_nontemporal_load((const v4f*)(row + e));
      const v4f w = *(const v4f*)(vs + e);
      sum += d.x * w.x + d.y * w.y + d.z * w.z + d.w * w.w;
    }
    // wave32 butterfly reduction
    #pragma unroll
    for (int off = 16; off > 0; off >>= 1)
      sum += __shfl_xor(sum, off, 32);
    if (lane == 0) energ[(size_t)b * L_ + l] = sum + cb;
  }
}

// ---------------------------------------------------------------------------
// Kernel 4: in-place softmax over L=2048, one block (256 threads) per batch
// ---------------------------------------------------------------------------
__global__ __launch_bounds__(256) void softmax_kernel(float* __restrict__ out)
{
  __shared__ float red[256];
  float* row = out + (size_t)blockIdx.x * L_;
  float vals[8];
  float m = -INFINITY;
  #pragma unroll
  for (int t = 0; t < 8; ++t) {
    vals[t] = row[threadIdx.x + 256 * t];
    m = fmaxf(m, vals[t]);
  }
  red[threadIdx.x] = m;
  __syncthreads();
  for (int off = 128; off > 0; off >>= 1) {
    if (threadIdx.x < off)
      red[threadIdx.x] = fmaxf(red[threadIdx.x], red[threadIdx.x + off]);
    __syncthreads();
  }
  m = red[0];
  __syncthreads();
  float s = 0.f;
  #pragma unroll
  for (int t = 0; t < 8; ++t) {
    vals[t] = __expf(vals[t] - m);
    s += vals[t];
  }
  red[threadIdx.x] = s;
  __syncthreads();
  for (int off = 128; off > 0; off >>= 1) {
    if (threadIdx.x < off) red[threadIdx.x] += red[threadIdx.x + off];
    __syncthreads();
  }
  const float inv = 1.f / red[0];
  #pragma unroll
  for (int t = 0; t < 8; ++t)
    row[threadIdx.x + 256 * t] = vals[t] * inv;
}

// ---------------------------------------------------------------------------
extern "C" void kernel_launch(void* const* d_in, const int* in_sizes, int n_in,
                              void* d_out, int out_size, void* d_ws, size_t ws_size,
                              hipStream_t stream)
{
  const float* hx   = (const float*)d_in[0];   // [32, 1024]
  const float* enc  = (const float*)d_in[1];   // [32, 2048, 1024]
  const float* W    = (const float*)d_in[2];   // [1024, 1024]
  const float* bias = (const float*)d_in[3];   // [1024]
  float* out = (float*)d_out;                  // [32, 1, 2048]

  float* v = (float*)d_ws;                     // 32*1024 floats = 128 KB
  float* c = v + (size_t)B_ * EN_;             // 32 floats

  proj_vec_wmma  <<<128,  32, 0, stream>>>(hx, W, v);
  bias_dot       <<<B_,  128, 0, stream>>>(hx, bias, c);
  energies_kernel<<<B_ * (L_ / 64), 256, 0, stream>>>(enc, v, c, out);
  softmax_kernel <<<B_,  256, 0, stream>>>(out);
}